// PointerGeneratorDistributionModule_54606214201414
// MI455X (gfx1250) — compile-verified
//
#include <hip/hip_runtime.h>
#include <hip/hip_bf16.h>
#include <stdint.h>

// ---------------------------------------------------------------------------
// Pointer-generator distribution, MI455X (gfx1250).
// out[t,b,e] = p_gen[t,b]*vocab[t,b,e] (e<V; 0 for e>=V)
//            + scatter_add over s of (1-p_gen[t,b])*attn[t,b,s] at ids[b,s].
// Pure bandwidth problem (~205 MB @ 23.3 TB/s). Data path: TDM load -> LDS,
// scale from LDS, non-temporal 128-bit stores. Scatter via f32 global atomics.
// ---------------------------------------------------------------------------

typedef float    f32x4    __attribute__((ext_vector_type(4)));
typedef unsigned uint32x4 __attribute__((ext_vector_type(4)));
typedef int      int32x8  __attribute__((ext_vector_type(8)));
typedef int      int32x4  __attribute__((ext_vector_type(4)));

#define CHUNK 2048   // floats per block tile (8 KB LDS)

__device__ __forceinline__ uint32_t lds_offset_u32(const void* p) {
  // Low 32 bits of a flat pointer into the LDS aperture are the LDS byte
  // offset (ISA 10.2 aperture mapping).
  return (uint32_t)(uintptr_t)p;
}

__global__ void __launch_bounds__(256)
pg_scale_pad_kernel(const float* __restrict__ p_gen,
                    const float* __restrict__ vocab,
                    float* __restrict__ out,
                    int V, int E, int nchunks) {
  __shared__ __align__(16) float tile[CHUNK];

  const int row   = blockIdx.x / nchunks;   // row = t*B + b
  const int chunk = blockIdx.x % nchunks;
  const int start = chunk * CHUNK;
  const int len   = min(CHUNK, V - start);  // 2048 or 848; both %4 == 0

  const float* gsrc = vocab + (size_t)row * (size_t)V + (size_t)start;

  // --- Wave 0 issues one TDM descriptor: 1-D tile, global -> LDS ---------
  // readfirstlane makes the condition wave-uniform in an SGPR, so waves 1..7
  // take a scalar branch around the TDM op (TDM ignores EXEC).
  if (__builtin_amdgcn_readfirstlane((int)threadIdx.x) < 32) {
    const uint64_t ga  = (uint64_t)(uintptr_t)gsrc;
    const uint32_t lds = lds_offset_u32(&tile[0]);

    uint32x4 g0;
    g0.x = 1u;                                  // count=1, user D#, no gather
    g0.y = lds;                                 // lds_addr (bytes)
    g0.z = (uint32_t)ga;                        // global_addr[31:0]
    g0.w = ((uint32_t)(ga >> 32) & 0x01FFFFFFu) // global_addr[56:32]
           | 0x80000000u;                       // type = 2 ("image")

    const uint32_t ulen = (uint32_t)len;
    int32x8 g1;
    g1[0] = (int)(2u << 16);                    // data_size = 4 bytes; no wg mask
    g1[1] = (int)((ulen & 0xFFFFu) << 16);      // tensor_dim0[15:0]  @ bits 63:48
    g1[2] = (int)((ulen >> 16) & 0xFFFFu);      // tensor_dim0[31:16]; tensor_dim1=0
    g1[3] = (int)((ulen & 0xFFFFu) << 16);      // tile_dim0          @ bits 127:112
    g1[4] = 0;                                  // tile_dim1=0, tile_dim2=0
    g1[5] = (int)ulen;                          // tensor_dim0_stride[31:0]
    g1[6] = 0;                                  // stride hi / tensor_dim1_stride lo
    g1[7] = 0;

    int32x4 g2 = {0, 0, 0, 0};                  // dims 2..4 unused (<=2D)
    int32x4 g3 = {0, 0, 0, 0};
    int32x8 g4 = {0, 0, 0, 0, 0, 0, 0, 0};      // clang-23 6-arg form extra group

    __builtin_amdgcn_tensor_load_to_lds(g0, g1, g2, g3, g4, /*cpol=*/0);
    __builtin_amdgcn_s_wait_tensorcnt(0);       // LDS tile is resident
  }
  __syncthreads();

  // --- All waves: scale row-uniform p_gen, stream out with NT b128 stores ---
  const float pg = p_gen[row];                  // uniform -> scalar load
  f32x4* __restrict__ out4 =
      (f32x4*)(out + (size_t)row * (size_t)E + (size_t)start);
  const f32x4* t4 = (const f32x4*)tile;
  const int n4 = len >> 2;

  for (int i = threadIdx.x; i < n4; i += blockDim.x) {
    f32x4 v = t4[i];                            // ds_load_b128
    v *= pg;
    __builtin_nontemporal_store(v, &out4[i]);   // global_store_b128, TH=NT
  }

  // Last chunk of the row also zero-fills the OOV pad region [V, E).
  if (chunk == nchunks - 1) {
    f32x4* pad4 = (f32x4*)(out + (size_t)row * (size_t)E + (size_t)V);
    const int npad4 = (E - V) >> 2;             // 16 for max_oov=64
    const f32x4 z = {0.f, 0.f, 0.f, 0.f};
    for (int i = threadIdx.x; i < npad4; i += blockDim.x)
      __builtin_nontemporal_store(z, &pad4[i]);
  }
}

__global__ void __launch_bounds__(256)
pg_scatter_kernel(const float* __restrict__ p_gen,
                  const float* __restrict__ attn,
                  const int* __restrict__ ids,
                  float* __restrict__ out,
                  int B, int S, int E, int total) {
  int idx = blockIdx.x * blockDim.x + threadIdx.x;
  if (idx >= total) return;                     // total = T*B*S
  const int s   = idx % S;
  const int tb  = idx / S;                      // tb = t*B + b
  const int b   = tb % B;
  const float w = (1.0f - p_gen[tb]) * attn[idx];
  const int col = ids[b * S + s];               // in [0, E)
  atomicAdd(out + (size_t)tb * (size_t)E + (size_t)col, w);  // global_atomic_add_f32
}

extern "C" void kernel_launch(void* const* d_in, const int* in_sizes, int n_in,
                              void* d_out, int out_size, void* d_ws, size_t ws_size,
                              hipStream_t stream) {
  const float* p_gen = (const float*)d_in[0];   // (T,B,1)
  const float* vocab = (const float*)d_in[1];   // (T,B,V)
  const float* attn  = (const float*)d_in[2];   // (T,B,S)
  const int*   ids   = (const int*)d_in[3];     // (B,S)
  // d_in[4] = oov_count (unused by the math), d_in[5] = max_oov scalar (unused)
  float* out = (float*)d_out;                   // (T,B,E)

  const int TB = in_sizes[0];                   // T*B
  const int V  = in_sizes[1] / TB;
  const int B  = in_sizes[4];
  const int S  = in_sizes[3] / B;
  const int E  = out_size / TB;                 // V + max_oov

  const int nchunks = (V + CHUNK - 1) / CHUNK;

  // Pass 1: out = pad(p_gen * vocab). Writes every element of d_out.
  pg_scale_pad_kernel<<<dim3(TB * nchunks), dim3(256), 0, stream>>>(
      p_gen, vocab, out, V, E, nchunks);

  // Pass 2: out += scatter((1 - p_gen) * attn). Same stream => ordered.
  const int total = TB * S;
  pg_scatter_kernel<<<dim3((total + 255) / 256), dim3(256), 0, stream>>>(
      p_gen, attn, ids, out, B, S, E, total);
}